// LowRankNufftOperator_1906965480023
// MI455X (gfx1250) — compile-verified
//
#include <hip/hip_runtime.h>

typedef __attribute__((ext_vector_type(2))) float v2f;
typedef __attribute__((ext_vector_type(8))) float v8f;

#define GX      512
#define NXIN    256
#define NPLANE  15          // (b*nslc*nc)*K collapsed per-plane count = 3*5
#define KRANK   5
#define NSLC    3
#define MSAMP   150000
#define NNZROW  80

// ---------------------------------------------------------------------------
// 1) DFT twiddle factor table W[u][n], u in [0,512), n in [0,256)
//    W = exp(-2*pi*i*u*n/512); periodic mod 512 to keep fp32 argument small.
// ---------------------------------------------------------------------------
__global__ __launch_bounds__(256)
void twiddle_kernel(float* __restrict__ Wr, float* __restrict__ Wi) {
  const int idx = blockIdx.x * 256 + threadIdx.x;       // exactly 512*256
  const int u = idx >> 8;
  const int n = idx & 255;
  const int t = (u * n) & 511;
  const float ang = -1.22718463e-2f * (float)t;         // -2*pi/512
  float s, c;
  __sincosf(ang, &s, &c);
  Wr[idx] = c;
  Wi[idx] = s;
}

// ---------------------------------------------------------------------------
// 2) Apodize: g[p] = apod * x[p]  (complex, 15 planes of 256x256)
// ---------------------------------------------------------------------------
__global__ __launch_bounds__(256)
void apod_kernel(const float* __restrict__ xr, const float* __restrict__ xi,
                 const float* __restrict__ ar, const float* __restrict__ ai,
                 float* __restrict__ gr, float* __restrict__ gi) {
  const int idx = blockIdx.x * 256 + threadIdx.x;       // 15*65536
  const int i = idx & 65535;
  const float a = ar[i], b = ai[i];
  const float xre = xr[idx], xim = xi[idx];
  gr[idx] = a * xre - b * xim;
  gi[idx] = a * xim + b * xre;
}

// ---------------------------------------------------------------------------
// 3) Stage-1 DFT as complex matmul via WMMA f32 16x16x4:
//    T[p][u][m] = sum_{n<256} W[u][n] * g[p][n][m]
//    One wave per 16x16 output tile; 64 K-chunks; EXEC uniform.
// ---------------------------------------------------------------------------
__global__ __launch_bounds__(128)
void dft_stage1_wmma(const float* __restrict__ Wr, const float* __restrict__ Wi,
                     const float* __restrict__ gr, const float* __restrict__ gi,
                     float* __restrict__ Tr, float* __restrict__ Ti) {
  const int tile = blockIdx.x * 4 + (threadIdx.x >> 5); // 15*32*16 = 7680 tiles
  const int lane = threadIdx.x & 31;
  const int hi = lane >> 4;                             // lane group: K+0/1 vs K+2/3
  const int l  = lane & 15;

  const int p  = tile >> 9;                             // / (32*16)
  const int r  = tile & 511;
  const int u0 = (r >> 4) << 4;
  const int m0 = (r & 15) << 4;

  const float* wrow_r  = Wr + (u0 + l) * NXIN;          // A row for this lane
  const float* wrow_i  = Wi + (u0 + l) * NXIN;
  const float* gcol_r  = gr + p * (NXIN * NXIN) + m0 + l;  // B col for this lane
  const float* gcol_i  = gi + p * (NXIN * NXIN) + m0 + l;

  v8f cr = {};
  v8f ci = {};

  for (int kk = 0; kk < 64; ++kk) {
    const int n = (kk << 2) + (hi << 1);
    v2f ar, ai, br, bi, ain;
    ar.x = wrow_r[n];            ar.y = wrow_r[n + 1];
    ai.x = wrow_i[n];            ai.y = wrow_i[n + 1];
    br.x = gcol_r[n * NXIN];     br.y = gcol_r[(n + 1) * NXIN];
    bi.x = gcol_i[n * NXIN];     bi.y = gcol_i[(n + 1) * NXIN];
    ain.x = -ai.x;               ain.y = -ai.y;
    // Cr += Ar*Br - Ai*Bi ; Ci += Ar*Bi + Ai*Br
    cr = __builtin_amdgcn_wmma_f32_16x16x4_f32(false, ar,  false, br, (short)0, cr, false, false);
    cr = __builtin_amdgcn_wmma_f32_16x16x4_f32(false, ain, false, bi, (short)0, cr, false, false);
    ci = __builtin_amdgcn_wmma_f32_16x16x4_f32(false, ar,  false, bi, (short)0, ci, false, false);
    ci = __builtin_amdgcn_wmma_f32_16x16x4_f32(false, ai,  false, br, (short)0, ci, false, false);
  }

  float* tr = Tr + p * (GX * NXIN) + m0 + l;
  float* ti = Ti + p * (GX * NXIN) + m0 + l;
#pragma unroll
  for (int j = 0; j < 8; ++j) {
    const int row = u0 + j + (hi << 3);                 // C layout: M = j + 8*hi
    tr[row * NXIN] = cr[j];
    ti[row * NXIN] = ci[j];
  }
}

// ---------------------------------------------------------------------------
// 4) Stage-2 DFT + scatter into L2-resident gather table:
//    Z[p][u][v] = sum_{m<256} T[p][u][m] * W[v][m]
//    table[n*6 + s*2 + {0,1}] = {Zr, Zi},  n = k*512*512 + u*512 + v,  p = s*5+k
// ---------------------------------------------------------------------------
__global__ __launch_bounds__(128)
void dft_stage2_wmma(const float* __restrict__ Wr, const float* __restrict__ Wi,
                     const float* __restrict__ Tr, const float* __restrict__ Ti,
                     float* __restrict__ table) {
  const int tile = blockIdx.x * 4 + (threadIdx.x >> 5); // 15*32*32 = 15360 tiles
  const int lane = threadIdx.x & 31;
  const int hi = lane >> 4;
  const int l  = lane & 15;

  const int p  = tile >> 10;                            // / (32*32)
  const int r  = tile & 1023;
  const int u0 = (r >> 5) << 4;
  const int v0 = (r & 31) << 4;

  const float* trow_r = Tr + p * (GX * NXIN) + (u0 + l) * NXIN;  // A row
  const float* trow_i = Ti + p * (GX * NXIN) + (u0 + l) * NXIN;
  const float* wcol_r = Wr + (v0 + l) * NXIN;           // B(k,col)=W[v0+col][m+k]
  const float* wcol_i = Wi + (v0 + l) * NXIN;

  v8f cr = {};
  v8f ci = {};

  for (int kk = 0; kk < 64; ++kk) {
    const int m = (kk << 2) + (hi << 1);
    v2f ar, ai, br, bi, ain;
    ar.x = trow_r[m];    ar.y = trow_r[m + 1];
    ai.x = trow_i[m];    ai.y = trow_i[m + 1];
    br.x = wcol_r[m];    br.y = wcol_r[m + 1];
    bi.x = wcol_i[m];    bi.y = wcol_i[m + 1];
    ain.x = -ai.x;       ain.y = -ai.y;
    cr = __builtin_amdgcn_wmma_f32_16x16x4_f32(false, ar,  false, br, (short)0, cr, false, false);
    cr = __builtin_amdgcn_wmma_f32_16x16x4_f32(false, ain, false, bi, (short)0, cr, false, false);
    ci = __builtin_amdgcn_wmma_f32_16x16x4_f32(false, ar,  false, bi, (short)0, ci, false, false);
    ci = __builtin_amdgcn_wmma_f32_16x16x4_f32(false, ai,  false, br, (short)0, ci, false, false);
  }

  const int s    = p / KRANK;
  const int ksub = p % KRANK;
  const int col  = v0 + l;
#pragma unroll
  for (int j = 0; j < 8; ++j) {
    const int row = u0 + j + (hi << 3);
    const long n = (long)ksub * (GX * GX) + (long)row * GX + col;
    float2 z;
    z.x = cr[j];
    z.y = ci[j];
    *(float2*)(table + n * 6 + s * 2) = z;              // 8B-aligned interleaved store
  }
}

// ---------------------------------------------------------------------------
// 5) Sparse complex interpolation + low-rank slice mixing + rdcf.
//    One wave32 per k-space sample m (rows are repeat(arange(M),80), sorted):
//    lanes stride the 80 nnz -> coalesced streams, L2-served gathers,
//    shuffle-XOR reduction, lanes 0..2 finish one slice each.
// ---------------------------------------------------------------------------
__global__ __launch_bounds__(256)
void sparse_mix_kernel(const float* __restrict__ vals_r, const float* __restrict__ vals_i,
                       const int* __restrict__ cols,
                       const float* __restrict__ phi_re, const float* __restrict__ phi_im,
                       const float* __restrict__ rdcf,
                       const float* __restrict__ table, float* __restrict__ out) {
  const int m    = (blockIdx.x * blockDim.x + threadIdx.x) >> 5;  // 150000 exactly
  const int lane = threadIdx.x & 31;
  const long base = (long)m * NNZROW;

  float a0r = 0.f, a0i = 0.f, a1r = 0.f, a1i = 0.f, a2r = 0.f, a2i = 0.f;

  for (int j = lane; j < NNZROW; j += 32) {
    const long idx = base + j;
    __builtin_prefetch(vals_r + idx + 960, 0, 0);       // stream ahead (global_prefetch_b8)
    __builtin_prefetch(cols + idx + 960, 0, 0);
    const float vr = vals_r[idx];
    const float vi = vals_i[idx];
    const int   c  = cols[idx];
    const float* t = table + (long)c * 6;               // 24B column, L2-resident
    float gr = t[0], gi = t[1];
    a0r += vr * gr - vi * gi;  a0i += vr * gi + vi * gr;
    gr = t[2]; gi = t[3];
    a1r += vr * gr - vi * gi;  a1i += vr * gi + vi * gr;
    gr = t[4]; gi = t[5];
    a2r += vr * gr - vi * gi;  a2i += vr * gi + vi * gr;
  }

  for (int off = 16; off; off >>= 1) {
    a0r += __shfl_xor(a0r, off, 32);  a0i += __shfl_xor(a0i, off, 32);
    a1r += __shfl_xor(a1r, off, 32);  a1i += __shfl_xor(a1i, off, 32);
    a2r += __shfl_xor(a2r, off, 32);  a2i += __shfl_xor(a2i, off, 32);
  }

  if (lane < NSLC) {
    const int s = lane;
    const float p0r = phi_re[0 * MSAMP + m], p0i = phi_im[0 * MSAMP + m];
    const float p1r = phi_re[1 * MSAMP + m], p1i = phi_im[1 * MSAMP + m];
    const float p2r = phi_re[2 * MSAMP + m], p2i = phi_im[2 * MSAMP + m];

    // S = sum_s A[s]*P[s]
    const float Sr = (a0r * p0r - a0i * p0i) + (a1r * p1r - a1i * p1i) + (a2r * p2r - a2i * p2i);
    const float Si = (a0r * p0i + a0i * p0r) + (a1r * p1i + a1i * p1r) + (a2r * p2i + a2i * p2r);

    const float ar_ = (s == 0) ? a0r : ((s == 1) ? a1r : a2r);
    const float ai_ = (s == 0) ? a0i : ((s == 1) ? a1i : a2i);
    const float pr  = (s == 0) ? p0r : ((s == 1) ? p1r : p2r);
    const float pi  = (s == 0) ? p0i : ((s == 1) ? p1i : p2i);

    // Q = S - A[s]*P[s] ; y = A[s] + conj(P[s])*Q ; y *= rdcf
    const float qr = Sr - (ar_ * pr - ai_ * pi);
    const float qi = Si - (ar_ * pi + ai_ * pr);
    const float w  = rdcf[m];
    const float yr = (ar_ + (pr * qr + pi * qi)) * w;
    const float yi = (ai_ + (pr * qi - pi * qr)) * w;

    out[((long)s * MSAMP + m) * 2 + 0] = yr;            // [b=1, nslc, M, nc=1, 2]
    out[((long)s * MSAMP + m) * 2 + 1] = yi;
  }
}

// ---------------------------------------------------------------------------
extern "C" void kernel_launch(void* const* d_in, const int* in_sizes, int n_in,
                              void* d_out, int out_size, void* d_ws, size_t ws_size,
                              hipStream_t stream) {
  const float* x_re    = (const float*)d_in[0];
  const float* x_im    = (const float*)d_in[1];
  const float* apod_re = (const float*)d_in[2];
  const float* apod_im = (const float*)d_in[3];
  const float* vals_r  = (const float*)d_in[4];
  const float* vals_i  = (const float*)d_in[5];
  const float* phi_re  = (const float*)d_in[6];
  const float* phi_im  = (const float*)d_in[7];
  const float* rdcf    = (const float*)d_in[8];
  // d_in[9] = rows: structurally repeat(arange(M), 80) -> not needed
  const int*   cols    = (const int*)d_in[10];
  float*       out     = (float*)d_out;

  // Workspace carve-up (floats): total ~56.1 MB
  float* ws    = (float*)d_ws;
  float* Wr    = ws;                              // 512*256
  float* Wi    = Wr + (GX * NXIN);
  float* gr    = Wi + (GX * NXIN);                // 15*256*256
  float* gi    = gr + (NPLANE * NXIN * NXIN);
  float* Tr    = gi + (NPLANE * NXIN * NXIN);     // 15*512*256
  float* Ti    = Tr + (NPLANE * GX * NXIN);
  float* table = Ti + (NPLANE * GX * NXIN);       // N*6 = 1310720*6

  twiddle_kernel<<<(GX * NXIN) / 256, 256, 0, stream>>>(Wr, Wi);
  apod_kernel<<<(NPLANE * NXIN * NXIN) / 256, 256, 0, stream>>>(
      x_re, x_im, apod_re, apod_im, gr, gi);
  dft_stage1_wmma<<<(NPLANE * 32 * 16) / 4, 128, 0, stream>>>(Wr, Wi, gr, gi, Tr, Ti);
  dft_stage2_wmma<<<(NPLANE * 32 * 32) / 4, 128, 0, stream>>>(Wr, Wi, Tr, Ti, table);
  sparse_mix_kernel<<<(MSAMP * 32) / 256, 256, 0, stream>>>(
      vals_r, vals_i, cols, phi_re, phi_im, rdcf, table, out);
}